// MPSROIPooling_30623116821372
// MI455X (gfx1250) — compile-verified
//
#include <hip/hip_runtime.h>
#include <math.h>
#include <stdint.h>

// ---------------------------------------------------------------------------
// ROI max pooling (Fast R-CNN, 7x7 bins) for MI455X / gfx1250.
// Data movement via the CDNA5 Tensor Data Mover: each ROI window is a strided
// 3D tile (roi_w x roi_h x CC channels) DMA'd into LDS with
// tensor_load_to_lds, double-buffered on TENSORcnt. Compute is a per-bin max
// reduction out of LDS. Channels are split across blocks (grid = R x C/32)
// to maximize WGP occupancy and shorten the per-block TDM dependency chain.
// ---------------------------------------------------------------------------

typedef __attribute__((ext_vector_type(4))) unsigned int v4u;
typedef __attribute__((ext_vector_type(4))) int          v4i;
typedef __attribute__((ext_vector_type(8))) int          v8i;

#define POOL_H 7
#define POOL_W 7
#define BDIM   128               // 4 wave32
#define CC     2                 // channels per TDM chunk
#define CPB    32                // channels per block (16 chunks, double-buffered)
#define MAXTILE (CC * 50 * 50)   // floats per LDS buffer (worst-case ROI = full map)

__device__ __forceinline__ unsigned rfl(unsigned x) {
  return (unsigned)__builtin_amdgcn_readfirstlane((int)x);
}

// Issue one TDM load: 3D tile (tile_w x tile_h x tile_c elements, 4B each)
// from global (row stride s0 elems, plane stride s1 elems) into LDS @ lds_addr.
__device__ __forceinline__ void tdm_load_tile(unsigned lds_addr,
                                              unsigned long long ga,
                                              int tile_w, int tile_h, int tile_c,
                                              unsigned td0, unsigned td1, unsigned td2,
                                              long long s0, long long s1) {
  v4u g0;
  g0[0] = rfl(1u);                                                   // count=1, user desc
  g0[1] = rfl(lds_addr);                                             // lds_addr (bytes)
  g0[2] = rfl((unsigned)ga);                                         // global_addr[31:0]
  g0[3] = rfl(((unsigned)(ga >> 32) & 0x01FFFFFFu) | 0x80000000u);   // ga[56:32] | type=2

  v8i g1;
  g1[0] = (int)rfl(2u << 16);                                        // data_size=4B, no mask
  g1[1] = (int)rfl((td0 & 0xFFFFu) << 16);                           // tensor_dim0 lo16
  g1[2] = (int)rfl(((td0 >> 16) & 0xFFFFu) | ((td1 & 0xFFFFu) << 16)); // td0 hi / td1 lo
  g1[3] = (int)rfl(((td1 >> 16) & 0xFFFFu) |
                   (((unsigned)tile_w & 0xFFFFu) << 16));            // td1 hi / tile_dim0
  g1[4] = (int)rfl(((unsigned)tile_h & 0xFFFFu) |
                   (((unsigned)tile_c & 0xFFFFu) << 16));            // tile_dim1 / tile_dim2
  g1[5] = (int)rfl((unsigned)(s0 & 0xFFFFFFFFll));                   // dim0_stride[31:0]
  g1[6] = (int)rfl((unsigned)((s0 >> 32) & 0xFFFFll) |
                   ((unsigned)(s1 & 0xFFFFll) << 16));               // dim0_str hi / dim1_str lo
  g1[7] = (int)rfl((unsigned)((s1 >> 16) & 0xFFFFFFFFll));           // dim1_stride[47:16]

  v4i g2 = { (int)rfl(td2), 0, 0, 0 };                               // tensor_dim2; 3D tile
  v4i g3 = { 0, 0, 0, 0 };

#if __clang_major__ >= 23
  v8i g4 = { 0, 0, 0, 0, 0, 0, 0, 0 };
  __builtin_amdgcn_tensor_load_to_lds(g0, g1, g2, g3, g4, 0);
#else
  __builtin_amdgcn_tensor_load_to_lds(g0, g1, g2, g3, 0);
#endif
}

__global__ void roi_max_pool_tdm(const float* __restrict__ fm,
                                 const float* __restrict__ rois,
                                 float* __restrict__ out,
                                 int B, int C, int H, int W) {
  extern __shared__ float smem[];

  const int r      = blockIdx.x;
  const int c_base = blockIdx.y * CPB;
  const int tid    = threadIdx.x;
  const float* roi = rois + (size_t)r * 5;

  // Uniform ROI geometry (all threads compute the same values).
  int b  = (int)roi[0];
  int x1 = (int)roundf(roi[1]);
  int y1 = (int)roundf(roi[2]);
  int x2 = (int)roundf(roi[3]);
  int y2 = (int)roundf(roi[4]);
  x1 = min(max(x1, 0), W - 1);
  y1 = min(max(y1, 0), H - 1);
  const int roi_w = min(max(x2 - x1 + 1, 1), W - x1);
  const int roi_h = min(max(y2 - y1 + 1, 1), H - y1);
  const float bin_w = (float)roi_w / (float)POOL_W;
  const float bin_h = (float)roi_h / (float)POOL_H;

  const unsigned lds_base  = (unsigned)(uintptr_t)smem;  // flat LDS low 32b = LDS offset
  const unsigned buf_bytes = (unsigned)(MAXTILE * sizeof(float));
  const unsigned long long fm_base =
      (unsigned long long)(uintptr_t)fm +
      4ull * ((unsigned long long)(b * C + c_base) * H * W +
              (unsigned long long)y1 * W + (unsigned long long)x1);
  const unsigned long long chunk_stride = 4ull * (unsigned long long)CC * H * W;

  const int nchunks = CPB / CC;     // 16
  const bool leader = (tid < 32);   // wave 0 issues the TDM ops

  if (leader) {
    tdm_load_tile(lds_base, fm_base, roi_w, roi_h, CC,
                  (unsigned)W, (unsigned)H, (unsigned)C,
                  (long long)W, (long long)H * W);
  }

  const int tile_elems = roi_h * roi_w;

  for (int k = 0; k < nchunks; ++k) {
    // Prefetch next chunk into the other buffer while this one finishes.
    if (leader && (k + 1) < nchunks) {
      tdm_load_tile(lds_base + (unsigned)((k + 1) & 1) * buf_bytes,
                    fm_base + (unsigned long long)(k + 1) * chunk_stride,
                    roi_w, roi_h, CC,
                    (unsigned)W, (unsigned)H, (unsigned)C,
                    (long long)W, (long long)H * W);
    }
    if ((k + 1) < nchunks) {
      __builtin_amdgcn_s_wait_tensorcnt(1);   // chunk k complete (in-order TDM)
    } else {
      __builtin_amdgcn_s_wait_tensorcnt(0);
    }
    __syncthreads();

    const float* buf = smem + (unsigned)(k & 1) * MAXTILE;

    // 98 bins (CC channels x 49 bins) for 128 threads: one bin per thread.
    for (int idx = tid; idx < CC * POOL_H * POOL_W; idx += BDIM) {
      const int c_local = idx / (POOL_H * POOL_W);
      const int bin     = idx % (POOL_H * POOL_W);
      const int ph = bin / POOL_W;
      const int pw = bin % POOL_W;

      int hs = (int)floorf((float)ph * bin_h);
      int he = (int)ceilf((float)(ph + 1) * bin_h);
      int ws = (int)floorf((float)pw * bin_w);
      int we = (int)ceilf((float)(pw + 1) * bin_w);
      hs = min(max(hs, 0), roi_h);
      he = min(max(he, 0), roi_h);
      ws = min(max(ws, 0), roi_w);
      we = min(max(we, 0), roi_w);

      float m = -INFINITY;
      const float* base = buf + c_local * tile_elems;
      for (int h = hs; h < he; ++h) {
        const float* row = base + h * roi_w;
        for (int w = ws; w < we; ++w) m = fmaxf(m, row[w]);
      }
      if (hs >= he || ws >= we) m = 0.0f;   // empty bin -> 0 (torchvision convention)

      const int c = c_base + k * CC + c_local;
      out[(((size_t)r * C + c) * POOL_H + ph) * POOL_W + pw] = m;
    }
    __syncthreads();  // compute on buf[k&1] done before chunk k+2 overwrites it
  }
}

extern "C" void kernel_launch(void* const* d_in, const int* in_sizes, int n_in,
                              void* d_out, int out_size, void* d_ws, size_t ws_size,
                              hipStream_t stream) {
  const float* fm   = (const float*)d_in[0];
  const float* rois = (const float*)d_in[1];
  float* out        = (float*)d_out;

  const int B = 2, C = 256, H = 50, W = 50;
  const int R = in_sizes[1] / 5;

  const size_t shmem = 2 * (size_t)MAXTILE * sizeof(float);  // 40 KB double buffer
  dim3 grid(R, C / CPB);                                     // 256 x 8 = 2048 workgroups
  roi_max_pool_tdm<<<grid, BDIM, shmem, stream>>>(fm, rois, out, B, C, H, W);
}